// GAT_25898652795239
// MI455X (gfx1250) — compile-verified
//
#include <hip/hip_runtime.h>
#include <math.h>

// ---------------------------------------------------------------------------
// GAT 2-layer forward for MI455X (gfx1250, wave32).
// GEMMs use V_WMMA_F32_16X16X4_F32 (fp32 tensor path -- workload is L2/atomic
// bound, so keep full precision). Edge softmax/aggregation via fp32 atomics.
// ---------------------------------------------------------------------------

typedef __attribute__((ext_vector_type(2))) float v2f;
typedef __attribute__((ext_vector_type(8))) float v8f;

#define NEG_SLOPE 0.2f

// Order-preserving map float -> uint so unsigned atomicMax == float max.
__device__ __forceinline__ unsigned f2ord(float f) {
    unsigned u = __float_as_uint(f);
    return (u & 0x80000000u) ? ~u : (u | 0x80000000u);
}
__device__ __forceinline__ float ord2f(unsigned o) {
    return (o & 0x80000000u) ? __uint_as_float(o & 0x7fffffffu)
                             : __uint_as_float(~o);
}

// ---------------------------------------------------------------------------
// C[nrows, NCOLS] = A[nrows, 128] @ W[128, NCOLS]   (row-major, fp32)
// One block = 16 output rows. Wave w owns columns [16w, 16w+16).
// A tile staged in LDS with pad-132 stride: bank = (4m + k) % 64, conflict-free.
// NCOLS compile-time -> shift-based addressing, no 64-bit mul in hot paths.
// ---------------------------------------------------------------------------
template <int NCOLS>
__global__ void gemm_wmma_k128(const float* __restrict__ A,
                               const float* __restrict__ W,
                               float* __restrict__ C,
                               int nrows) {
    __shared__ float As[16 * 132];
    const int lane = threadIdx.x & 31;
    const int wave = threadIdx.x >> 5;
    const int row0 = blockIdx.x * 16;

    // cooperative load of A tile [16 x 128] into LDS
    for (int i = threadIdx.x; i < 16 * 128; i += blockDim.x) {
        int r = i >> 7;
        int k = i & 127;
        int g = row0 + r;
        As[r * 132 + k] = (g < nrows) ? A[(size_t)g * 128 + k] : 0.0f;
    }
    __syncthreads();

    const int col0  = wave * 16;
    const int m     = lane & 15;          // A row within tile
    const int khalf = (lane >> 4) * 2;    // VGPR0 holds K=khalf, VGPR1 K=khalf+1
    const int n     = lane & 15;          // output column within tile

    const float* wp = W + (size_t)khalf * NCOLS + col0 + n;

    v8f acc = {0.f, 0.f, 0.f, 0.f, 0.f, 0.f, 0.f, 0.f};
    #pragma unroll 4
    for (int k0 = 0; k0 < 128; k0 += 4) {
        v2f a, b;
        a.x = As[m * 132 + k0 + khalf];
        a.y = As[m * 132 + k0 + khalf + 1];
        b.x = wp[(size_t)k0 * NCOLS];
        b.y = wp[(size_t)(k0 + 1) * NCOLS];
        acc = __builtin_amdgcn_wmma_f32_16x16x4_f32(
            /*neg_a=*/false, a, /*neg_b=*/false, b,
            /*c_mod=*/(short)0, acc, /*reuse_a=*/false, /*reuse_b=*/false);
    }

    // D layout: VGPR r -> M = r (lanes 0-15) / r+8 (lanes 16-31); N = lane&15
    const int mbase = (lane >> 4) * 8;
    float* cp = C + (size_t)(row0 + mbase) * NCOLS + col0 + n;
    if (row0 + 16 <= nrows) {
        // uniform fast path: whole tile in-bounds (N % 16 == 0 here), no guards
        #pragma unroll
        for (int r = 0; r < 8; ++r) cp[(size_t)r * NCOLS] = acc[r];
    } else {
        #pragma unroll
        for (int r = 0; r < 8; ++r)
            if (row0 + mbase + r < nrows) cp[(size_t)r * NCOLS] = acc[r];
    }
}

// a_src[n,h] = sum_c h[n,h,c]*att_src[h,c] ; same for a_dst
__global__ void attn_scores(const float* __restrict__ h,
                            const float* __restrict__ att_src,
                            const float* __restrict__ att_dst,
                            float* __restrict__ a_src,
                            float* __restrict__ a_dst,
                            int n, int H, int Cc) {
    int t = blockIdx.x * blockDim.x + threadIdx.x;
    if (t >= n * H) return;
    int node = t / H, hd = t - node * H;
    const float* hp = h + (size_t)node * H * Cc + hd * Cc;
    const float* as = att_src + hd * Cc;
    const float* ad = att_dst + hd * Cc;
    float s = 0.f, d = 0.f;
    for (int c = 0; c < Cc; ++c) { float v = hp[c]; s += v * as[c]; d += v * ad[c]; }
    a_src[t] = s;
    a_dst[t] = d;
}

__device__ __forceinline__ void edge_sd(const int* ei, int E, int e, int& s, int& d) {
    if (e < E) { s = ei[e]; d = ei[E + e]; }           // [2,E] flat
    else       { s = e - E; d = e - E; }               // self loop
}

// pass 1: segment max of leaky_relu(a_src[src]+a_dst[dst]) over dst
__global__ void edge_max(const int* __restrict__ ei, int E, int n, int H,
                         const float* __restrict__ a_src,
                         const float* __restrict__ a_dst,
                         unsigned* __restrict__ mord) {
    int e = blockIdx.x * blockDim.x + threadIdx.x;
    if (e >= E + n) return;
    int s, d; edge_sd(ei, E, e, s, d);
    for (int hd = 0; hd < H; ++hd) {
        float v = a_src[s * H + hd] + a_dst[d * H + hd];
        v = (v > 0.f) ? v : NEG_SLOPE * v;
        atomicMax(&mord[d * H + hd], f2ord(v));
    }
}

// pass 2: segment sum of exp(e - max)
__global__ void edge_denom(const int* __restrict__ ei, int E, int n, int H,
                           const float* __restrict__ a_src,
                           const float* __restrict__ a_dst,
                           const unsigned* __restrict__ mord,
                           float* __restrict__ den) {
    int e = blockIdx.x * blockDim.x + threadIdx.x;
    if (e >= E + n) return;
    int s, d; edge_sd(ei, E, e, s, d);
    for (int hd = 0; hd < H; ++hd) {
        float v = a_src[s * H + hd] + a_dst[d * H + hd];
        v = (v > 0.f) ? v : NEG_SLOPE * v;
        atomicAdd(&den[d * H + hd], expf(v - ord2f(mord[d * H + hd])));
    }
}

// pass 3: agg[dst] += alpha * h[src]; one thread per (edge, 4-channel quad)
__global__ void edge_aggregate(const int* __restrict__ ei, int E, int n,
                               int H, int Cc,
                               const float* __restrict__ a_src,
                               const float* __restrict__ a_dst,
                               const unsigned* __restrict__ mord,
                               const float* __restrict__ den,
                               const float* __restrict__ h,
                               float* __restrict__ agg) {
    const int CH  = H * Cc;
    const int qpe = CH >> 2;                       // quads per edge
    long long t = (long long)blockIdx.x * blockDim.x + threadIdx.x;
    if (t >= (long long)(E + n) * qpe) return;
    int e  = (int)(t / qpe);
    int c0 = (int)(t - (long long)e * qpe) * 4;
    int hd = c0 / Cc;                              // Cc % 4 == 0, quad stays in-head
    int s, d; edge_sd(ei, E, e, s, d);
    float v = a_src[s * H + hd] + a_dst[d * H + hd];
    v = (v > 0.f) ? v : NEG_SLOPE * v;
    float alpha = expf(v - ord2f(mord[d * H + hd])) / (den[d * H + hd] + 1e-16f);
    const float4 hv = *(const float4*)(h + (size_t)s * CH + c0);
    float* ap = agg + (size_t)d * CH + c0;
    atomicAdd(ap + 0, hv.x * alpha);
    atomicAdd(ap + 1, hv.y * alpha);
    atomicAdd(ap + 2, hv.z * alpha);
    atomicAdd(ap + 3, hv.w * alpha);
}

// out = relu(agg + bias)
__global__ void bias_relu(const float* __restrict__ agg,
                          const float* __restrict__ b,
                          float* __restrict__ out, int total, int CH) {
    int t = blockIdx.x * blockDim.x + threadIdx.x;
    if (t >= total) return;
    float v = agg[t] + b[t % CH];
    out[t] = v > 0.f ? v : 0.f;
}

// final: log_softmax(agg + b2) over 16 classes, one thread per node
__global__ void logsoftmax16(const float* __restrict__ agg,
                             const float* __restrict__ b,
                             float* __restrict__ out, int n) {
    int node = blockIdx.x * blockDim.x + threadIdx.x;
    if (node >= n) return;
    float v[16];
    float mx = -INFINITY;
    #pragma unroll
    for (int c = 0; c < 16; ++c) {
        v[c] = agg[(size_t)node * 16 + c] + b[c];
        mx = fmaxf(mx, v[c]);
    }
    float sum = 0.f;
    #pragma unroll
    for (int c = 0; c < 16; ++c) sum += expf(v[c] - mx);
    float lse = mx + logf(sum);
    #pragma unroll
    for (int c = 0; c < 16; ++c) out[(size_t)node * 16 + c] = v[c] - lse;
}

static inline unsigned cdiv(long long a, long long b) { return (unsigned)((a + b - 1) / b); }

extern "C" void kernel_launch(void* const* d_in, const int* in_sizes, int n_in,
                              void* d_out, int out_size, void* d_ws, size_t ws_size,
                              hipStream_t stream) {
    const float* x        = (const float*)d_in[0];
    const int*   ei       = (const int*)d_in[1];
    const float* W1       = (const float*)d_in[2];
    const float* att_src1 = (const float*)d_in[3];
    const float* att_dst1 = (const float*)d_in[4];
    const float* b1       = (const float*)d_in[5];
    const float* W2       = (const float*)d_in[6];
    const float* att_src2 = (const float*)d_in[7];
    const float* att_dst2 = (const float*)d_in[8];
    const float* b2       = (const float*)d_in[9];

    const int n    = in_sizes[0] / 128;   // N nodes (IN_CH = 128)
    const int E    = in_sizes[1] / 2;     // edges (edge_index is [2,E])
    const int Etot = E + n;               // + self loops

    // Workspace layout (floats), total 272*N ~= 54.4 MB
    float*    ws    = (float*)d_ws;
    float*    h1    = ws;                          // 128N ; reused as relu(h) input to L2
    float*    agg1  = ws + (size_t)128 * n;        // 128N ; L2 reuse: h2 (16N) + agg2 (16N)
    float*    asrc1 = ws + (size_t)256 * n;        // 4N
    float*    adst1 = asrc1 + (size_t)4 * n;       // 4N
    unsigned* m1    = (unsigned*)(adst1 + (size_t)4 * n);  // 4N
    float*    den1  = (float*)(m1 + (size_t)4 * n);        // 4N
    float*    h2    = agg1;                        // 16N (after agg1 consumed)
    float*    agg2  = agg1 + (size_t)16 * n;       // 16N
    float*    asrc2 = asrc1;                       // N
    float*    adst2 = adst1;                       // N
    unsigned* m2    = m1;                          // N
    float*    den2  = den1;                        // N

    // ---------------- layer 1 ----------------
    hipMemsetAsync(agg1, 0, (size_t)128 * n * sizeof(float), stream);
    hipMemsetAsync(m1,   0, (size_t)8  * n * sizeof(float), stream);  // m1 + den1

    gemm_wmma_k128<128><<<cdiv(n, 16), 256, 0, stream>>>(x, W1, h1, n);
    attn_scores<<<cdiv((long long)n * 4, 256), 256, 0, stream>>>(
        h1, att_src1, att_dst1, asrc1, adst1, n, 4, 32);
    edge_max<<<cdiv(Etot, 256), 256, 0, stream>>>(ei, E, n, 4, asrc1, adst1, m1);
    edge_denom<<<cdiv(Etot, 256), 256, 0, stream>>>(ei, E, n, 4, asrc1, adst1, m1, den1);
    edge_aggregate<<<cdiv((long long)Etot * 32, 256), 256, 0, stream>>>(
        ei, E, n, 4, 32, asrc1, adst1, m1, den1, h1, agg1);
    bias_relu<<<cdiv((long long)n * 128, 256), 256, 0, stream>>>(
        agg1, b1, h1, n * 128, 128);

    // ---------------- layer 2 ----------------
    hipMemsetAsync(agg2, 0, (size_t)16 * n * sizeof(float), stream);
    hipMemsetAsync(m1,   0, (size_t)8  * n * sizeof(float), stream);  // reuse as m2+den2

    gemm_wmma_k128<16><<<cdiv(n, 16), 32, 0, stream>>>(h1, W2, h2, n);
    attn_scores<<<cdiv(n, 256), 256, 0, stream>>>(
        h2, att_src2, att_dst2, asrc2, adst2, n, 1, 16);
    edge_max<<<cdiv(Etot, 256), 256, 0, stream>>>(ei, E, n, 1, asrc2, adst2, m2);
    edge_denom<<<cdiv(Etot, 256), 256, 0, stream>>>(ei, E, n, 1, asrc2, adst2, m2, den2);
    edge_aggregate<<<cdiv((long long)Etot * 4, 256), 256, 0, stream>>>(
        ei, E, n, 1, 16, asrc2, adst2, m2, den2, h2, agg2);

    logsoftmax16<<<cdiv(n, 256), 256, 0, stream>>>(agg2, b2, (float*)d_out, n);
}